// RecomNet_74217034875112
// MI455X (gfx1250) — compile-verified
//
#include <hip/hip_runtime.h>

#define N_NODES 50000
#define IN_DIM  128
#define HID     200
#define N_EDGES 800000
#define NTILES  13          // ceil(200/16) column tiles

typedef float v2f __attribute__((ext_vector_type(2)));
typedef float v8f __attribute__((ext_vector_type(8)));

// ---------------------------------------------------------------------------
// Weight pre-pack: W[K x 200] -> Wp[tile][ks2][lane][4], zero-padded to 208
// cols, laid out in exact V_WMMA_F32_16X16X4_F32 B-fragment lane order so the
// GEMM inner loop does ONE b128 load per two K-steps, branch-free.
//   lane = half*16 + lid ; B fragment for k-step ks: {W[4ks+2h][col], W[4ks+2h+1][col]}
//   float4 packs k-steps (2*ks2) and (2*ks2+1).
// ---------------------------------------------------------------------------
template <int K>
__global__ __launch_bounds__(256) void pack_W(const float* __restrict__ W,
                                              float* __restrict__ Wp)
{
    const int t = blockIdx.x * 256 + threadIdx.x;
    const int total = NTILES * (K / 8) * 32;
    if (t >= total) return;
    const int lane = t & 31;
    const int rest = t >> 5;
    const int ks2  = rest % (K / 8);
    const int tile = rest / (K / 8);
    const int half = lane >> 4;
    const int lid  = lane & 15;
    const int col  = tile * 16 + lid;
    const bool ok  = (col < HID);

    float4 v;
    {
        const int ka = (2 * ks2) * 4 + 2 * half;
        v.x = ok ? W[(size_t)(ka + 0) * HID + col] : 0.0f;
        v.y = ok ? W[(size_t)(ka + 1) * HID + col] : 0.0f;
    }
    {
        const int ka = (2 * ks2 + 1) * 4 + 2 * half;
        v.z = ok ? W[(size_t)(ka + 0) * HID + col] : 0.0f;
        v.w = ok ? W[(size_t)(ka + 1) * HID + col] : 0.0f;
    }
    ((float4*)Wp)[(size_t)(tile * (K / 8) + ks2) * 32 + lane] = v;
}

// ---------------------------------------------------------------------------
// GEMM: out = X[N_NODES x K] @ W[K x HID] + bias (+ msim + mrat, + relu).
// One wave32 -> one 16x16 tile via V_WMMA_F32_16X16X4_F32, branch-free loop.
// ---------------------------------------------------------------------------
template <int K, bool MSGS, bool RELU>
__global__ __launch_bounds__(256) void gemm_wmma_f32(
    const float* __restrict__ X, const float* __restrict__ Wp,
    const float* __restrict__ bias,
    const float* __restrict__ msim, const float* __restrict__ mrat,
    float* __restrict__ out)
{
    const int lane = threadIdx.x & 31;
    const int wave = threadIdx.x >> 5;
    const int half = lane >> 4;
    const int lid  = lane & 15;

    const int row_tile = blockIdx.y * 8 + wave;
    if (row_tile >= (N_NODES / 16)) return;   // wave-uniform
    const int row0 = row_tile * 16;
    const int col0 = blockIdx.x * 16;
    const int col  = col0 + lid;
    const bool col_ok = (col < HID);

    const float* __restrict__ xrow = X + (size_t)(row0 + lid) * K + 2 * half;
    const float4* __restrict__ wp =
        (const float4*)Wp + (size_t)blockIdx.x * (K / 8) * 32 + lane;

    v8f acc = {};
    #pragma unroll 4
    for (int s = 0; s < K / 8; ++s) {
        __builtin_prefetch(xrow + 8 * s + 64, 0, 0);     // global_prefetch_b8
        v2f a0 = *(const v2f*)(xrow + 8 * s);            // k-step 2s
        v2f a1 = *(const v2f*)(xrow + 8 * s + 4);        // k-step 2s+1
        float4 b4 = wp[(size_t)s * 32];                  // one b128 -> 2 B frags
        v2f b0 = {b4.x, b4.y};
        v2f b1 = {b4.z, b4.w};
        acc = __builtin_amdgcn_wmma_f32_16x16x4_f32(false, a0, false, b0,
                                                    (short)0, acc, false, false);
        acc = __builtin_amdgcn_wmma_f32_16x16x4_f32(false, a1, false, b1,
                                                    (short)0, acc, false, false);
    }

    // Epilogue: clamp column for loads, predicate only the stores.
    const int colc = col_ok ? col : (HID - 1);
    const float bv = bias[colc];
    #pragma unroll
    for (int v = 0; v < 8; ++v) {
        const int row = row0 + v + 8 * half;
        const size_t idx = (size_t)row * HID + colc;
        float r = acc[v] + bv;
        if (MSGS) r += msim[idx] + mrat[idx];
        if (RELU) r = fmaxf(r, 0.0f);
        if (col_ok) out[idx] = r;
    }
}

// ---------------------------------------------------------------------------
// Edge scatter: m[dst] += h[src], 8 channels/thread, native f32 L2 atomics.
// ---------------------------------------------------------------------------
#define CH_CHUNKS 25   // 200 / 8

__global__ __launch_bounds__(256) void scatter_add_edges(
    const int* __restrict__ edges,   // [2, N_EDGES]: row0 = src, row1 = dst
    const float* __restrict__ h, float* __restrict__ m)
{
    const long long t = (long long)blockIdx.x * blockDim.x + threadIdx.x;
    if (t >= (long long)N_EDGES * CH_CHUNKS) return;
    const int e  = (int)(t / CH_CHUNKS);
    const int c0 = (int)(t % CH_CHUNKS) * 8;

    const int src = edges[e];
    const int dst = edges[N_EDGES + e];

    const float* __restrict__ hs = h + (size_t)src * HID + c0;
    float* __restrict__ md = m + (size_t)dst * HID + c0;

    const float4 v0 = *(const float4*)(hs);
    const float4 v1 = *(const float4*)(hs + 4);

    __hip_atomic_fetch_add(md + 0, v0.x, __ATOMIC_RELAXED, __HIP_MEMORY_SCOPE_AGENT);
    __hip_atomic_fetch_add(md + 1, v0.y, __ATOMIC_RELAXED, __HIP_MEMORY_SCOPE_AGENT);
    __hip_atomic_fetch_add(md + 2, v0.z, __ATOMIC_RELAXED, __HIP_MEMORY_SCOPE_AGENT);
    __hip_atomic_fetch_add(md + 3, v0.w, __ATOMIC_RELAXED, __HIP_MEMORY_SCOPE_AGENT);
    __hip_atomic_fetch_add(md + 4, v1.x, __ATOMIC_RELAXED, __HIP_MEMORY_SCOPE_AGENT);
    __hip_atomic_fetch_add(md + 5, v1.y, __ATOMIC_RELAXED, __HIP_MEMORY_SCOPE_AGENT);
    __hip_atomic_fetch_add(md + 6, v1.z, __ATOMIC_RELAXED, __HIP_MEMORY_SCOPE_AGENT);
    __hip_atomic_fetch_add(md + 7, v1.w, __ATOMIC_RELAXED, __HIP_MEMORY_SCOPE_AGENT);
}

__global__ __launch_bounds__(256) void zero_f32x4(float4* __restrict__ p, int n4)
{
    const int i = blockIdx.x * blockDim.x + threadIdx.x;
    if (i < n4) p[i] = make_float4(0.f, 0.f, 0.f, 0.f);
}

extern "C" void kernel_launch(void* const* d_in, const int* in_sizes, int n_in,
                              void* d_out, int out_size, void* d_ws, size_t ws_size,
                              hipStream_t stream)
{
    const int*   edge_sim = (const int*)  d_in[0];
    const int*   edge_rat = (const int*)  d_in[1];
    const float* x        = (const float*)d_in[2];
    const float* W_self1  = (const float*)d_in[3];
    const float* b_self1  = (const float*)d_in[4];
    const float* W_sim1   = (const float*)d_in[5];
    const float* b_sim1   = (const float*)d_in[6];
    const float* W_rat1   = (const float*)d_in[7];
    const float* b_rat1   = (const float*)d_in[8];
    const float* W_self2  = (const float*)d_in[9];
    const float* b_self2  = (const float*)d_in[10];
    const float* W_sim2   = (const float*)d_in[11];
    const float* b_sim2   = (const float*)d_in[12];
    const float* W_rat2   = (const float*)d_in[13];
    const float* b_rat2   = (const float*)d_in[14];
    float* out = (float*)d_out;

    const size_t BUF = (size_t)N_NODES * HID * sizeof(float);     // 40 MB
    const size_t PK1 = (size_t)NTILES * (IN_DIM / 8) * 128;       // floats
    const size_t PK2 = (size_t)NTILES * (HID / 8) * 128;          // floats
    char* ws = (char*)d_ws;
    float* h_sim = (float*)(ws + 0 * BUF);
    float* h_rat = (float*)(ws + 1 * BUF);
    float* m_sim = (float*)(ws + 2 * BUF);
    float* m_rat = (float*)(ws + 3 * BUF);
    float* h1    = (float*)(ws + 4 * BUF);
    float* wpk   = (float*)(ws + 5 * BUF);
    float* wp_sim1  = wpk;
    float* wp_rat1  = wp_sim1 + PK1;
    float* wp_self1 = wp_rat1 + PK1;
    float* wp_sim2  = wp_self1 + PK1;
    float* wp_rat2  = wp_sim2 + PK2;
    float* wp_self2 = wp_rat2 + PK2;

    const dim3 gB(256);
    const dim3 gGemm(NTILES, (N_NODES / 16 + 7) / 8);   // 13 x 391, 8 waves/blk
    const int  nScatter = (int)(((long long)N_EDGES * CH_CHUNKS + 255) / 256);
    const int  n4 = N_NODES * HID / 4;
    const int  nZ = (n4 + 255) / 256;
    const int  nP1 = (NTILES * (IN_DIM / 8) * 32 + 255) / 256;
    const int  nP2 = (NTILES * (HID / 8) * 32 + 255) / 256;

    // ---- Weight packing (tiny, once per launch) ----
    pack_W<IN_DIM><<<nP1, gB, 0, stream>>>(W_sim1,  wp_sim1);
    pack_W<IN_DIM><<<nP1, gB, 0, stream>>>(W_rat1,  wp_rat1);
    pack_W<IN_DIM><<<nP1, gB, 0, stream>>>(W_self1, wp_self1);
    pack_W<HID>  <<<nP2, gB, 0, stream>>>(W_sim2,  wp_sim2);
    pack_W<HID>  <<<nP2, gB, 0, stream>>>(W_rat2,  wp_rat2);
    pack_W<HID>  <<<nP2, gB, 0, stream>>>(W_self2, wp_self2);

    // ---- Layer 1 ----
    gemm_wmma_f32<IN_DIM, false, false><<<gGemm, gB, 0, stream>>>(x, wp_sim1, b_sim1, nullptr, nullptr, h_sim);
    gemm_wmma_f32<IN_DIM, false, false><<<gGemm, gB, 0, stream>>>(x, wp_rat1, b_rat1, nullptr, nullptr, h_rat);
    zero_f32x4<<<nZ, gB, 0, stream>>>((float4*)m_sim, n4);
    zero_f32x4<<<nZ, gB, 0, stream>>>((float4*)m_rat, n4);
    scatter_add_edges<<<nScatter, gB, 0, stream>>>(edge_sim, h_sim, m_sim);
    scatter_add_edges<<<nScatter, gB, 0, stream>>>(edge_rat, h_rat, m_rat);
    gemm_wmma_f32<IN_DIM, true, true><<<gGemm, gB, 0, stream>>>(x, wp_self1, b_self1, m_sim, m_rat, h1);

    // ---- Layer 2 ----
    gemm_wmma_f32<HID, false, false><<<gGemm, gB, 0, stream>>>(h1, wp_sim2, b_sim2, nullptr, nullptr, h_sim);
    gemm_wmma_f32<HID, false, false><<<gGemm, gB, 0, stream>>>(h1, wp_rat2, b_rat2, nullptr, nullptr, h_rat);
    zero_f32x4<<<nZ, gB, 0, stream>>>((float4*)m_sim, n4);
    zero_f32x4<<<nZ, gB, 0, stream>>>((float4*)m_rat, n4);
    scatter_add_edges<<<nScatter, gB, 0, stream>>>(edge_sim, h_sim, m_sim);
    scatter_add_edges<<<nScatter, gB, 0, stream>>>(edge_rat, h_rat, m_rat);
    gemm_wmma_f32<HID, true, false><<<gGemm, gB, 0, stream>>>(h1, wp_self2, b_self2, m_sim, m_rat, out);
}